// Sampler_77575699300381
// MI455X (gfx1250) — compile-verified
//
#include <hip/hip_runtime.h>
#include <cstdint>
#include <cstddef>

#define BB 128
#define VV 128000
#define NB1 2048          // level-1 buckets over d in [-32, 0]
#define NB2 512           // level-2 sub-buckets
#define PL1 64            // buckets per lane, level 1 (2048/32)
#define PL2 16            // buckets per lane, level 2 (512/32)
#define NTHREADS 1024
#define NWAVES 32
#define W1 0.015625f      // 32/2048
#define INV_W1 64.0f
#define W2 (1.0f/32768.0f)
#define INV_W2 32768.0f
#define SC1 268435456.0f         // 2^28 fixed-point scale, level 1
#define SC2 17179869184.0f       // 2^34 fixed-point scale, level 2
#define CNT_SHIFT 46
#define MASK46 ((1ull << 46) - 1ull)
#define FMIN_F32 (-3.4028234663852886e38f)

__device__ __forceinline__ int bucket1(float d) {
  int b = (int)floorf((d + 32.0f) * INV_W1);
  return b < 0 ? 0 : (b > NB1 - 1 ? NB1 - 1 : b);
}
__device__ __forceinline__ int bucket2(float d, float lo) {
  int s = (int)floorf((d - lo) * INV_W2);
  return s < 0 ? 0 : (s > NB2 - 1 ? NB2 - 1 : s);
}
__device__ __forceinline__ float comp4(const float4& x, int c) {
  return c == 0 ? x.x : (c == 1 ? x.y : (c == 2 ? x.z : x.w));
}

// ---- deterministic block reductions (fixed shfl-tree order) --------------
__device__ float blockMax(float v, volatile float* red) {
  for (int o = 16; o > 0; o >>= 1) v = fmaxf(v, __shfl_down(v, o, 32));
  int lane = threadIdx.x & 31, w = threadIdx.x >> 5;
  __syncthreads();
  if (lane == 0) red[w] = v;
  __syncthreads();
  if (threadIdx.x < 32) {
    v = red[threadIdx.x];
    for (int o = 16; o > 0; o >>= 1) v = fmaxf(v, __shfl_down(v, o, 32));
    if (threadIdx.x == 0) red[0] = v;
  }
  __syncthreads();
  float r = red[0];
  __syncthreads();
  return r;
}
__device__ float blockSum(float v, volatile float* red) {
  for (int o = 16; o > 0; o >>= 1) v += __shfl_down(v, o, 32);
  int lane = threadIdx.x & 31, w = threadIdx.x >> 5;
  __syncthreads();
  if (lane == 0) red[w] = v;
  __syncthreads();
  if (threadIdx.x < 32) {
    v = red[threadIdx.x];
    for (int o = 16; o > 0; o >>= 1) v += __shfl_down(v, o, 32);
    if (threadIdx.x == 0) red[0] = v;
  }
  __syncthreads();
  float r = red[0];
  __syncthreads();
  return r;
}

// ---- wave-parallel descending-order crossing search ----------------------
__device__ __forceinline__ unsigned long long waveInclScan(unsigned long long x) {
  int lane = threadIdx.x & 31;
  for (int o = 1; o < 32; o <<= 1) {
    unsigned long long y = __shfl_up(x, o, 32);
    if (lane >= o) x += y;
  }
  return x;
}

// Executed by lanes 0..31. Scans weights w(b) for b = n-1 .. 0 (descending),
// finds first non-empty bucket where cumulative >= tgt. Returns bucket and
// cumulative-before-bucket (broadcast to all lanes).
template <typename F>
__device__ __forceinline__ bool waveFindCross(F w, int n, int perLane,
                                              unsigned long long tgt,
                                              int& bOut, unsigned long long& prevOut) {
  int lane = threadIdx.x & 31;
  int base = n - 1 - lane * perLane;
  unsigned long long lsum = 0ull;
  for (int i = 0; i < perLane; ++i) lsum += w(base - i);
  unsigned long long incl = waveInclScan(lsum);
  unsigned long long pre = incl - lsum;
  bool inSeg = (pre < tgt) && (incl >= tgt) && (lsum > 0ull);
  unsigned long long mask = __ballot(inSeg);
  if (mask == 0ull) return false;
  int src = __ffsll(mask) - 1;
  int fb = -1; unsigned long long prev = 0ull;
  if (lane == src) {
    unsigned long long cum = pre;
    for (int i = 0; i < perLane; ++i) {
      unsigned long long wi = w(base - i);
      if (wi > 0ull && cum + wi >= tgt) { fb = base - i; prev = cum; break; }
      cum += wi;
    }
  }
  bOut = __shfl(fb, src, 32);
  prevOut = __shfl(prev, src, 32);
  return true;
}

__global__ __launch_bounds__(NTHREADS) void Sampler_77575699300381_kernel(
    const float* __restrict__ logits, const float* __restrict__ temps,
    const int* __restrict__ topks, const float* __restrict__ topps,
    const float* __restrict__ topps2, const float* __restrict__ minps,
    const float* __restrict__ us,
    int* __restrict__ tok_out, float* __restrict__ lp_out, float* __restrict__ ntlp_out)
{
  __shared__ float red[32];
  __shared__ unsigned long long hist[NB1];      // packed: count<<46 | mass*2^28
  __shared__ unsigned long long subm[3][NB2];   // level-2 masses *2^34
  __shared__ unsigned scnt[NB2];                // level-2 counts (top-k)
  __shared__ __align__(16) float stage[2][NWAVES][128];
  __shared__ int s_cb[4];
  __shared__ unsigned long long s_cr[4];
  __shared__ int s_found[4];
  __shared__ unsigned long long s_ckr;
  __shared__ float s_thr[3];
  __shared__ float c_tp2, c_ts;
  __shared__ int c_s3;
  __shared__ unsigned long long selKey;

  const int row = blockIdx.x;
  const int tid = threadIdx.x;
  const float invT = 1.0f / temps[row];
  const float topp = topps[row], topp2 = topps2[row];
  const float minp = minps[row], uu = us[row];
  const int k = topks[row];
  const float* rowp = logits + (size_t)row * VV;
  const float4* r4 = (const float4*)rowp;

  // ---------------- P0: row max -------------------------------------------
  float mraw = FMIN_F32;
  for (int i = tid; i < VV / 4; i += NTHREADS) {
    float4 x = r4[i];
    mraw = fmaxf(mraw, fmaxf(fmaxf(x.x, x.y), fmaxf(x.z, x.w)));
  }
  const float m = blockMax(mraw, red) * invT;

  // ---------------- P1: packed histogram via async LDS staging ------------
  for (int i = tid; i < NB1; i += NTHREADS) hist[i] = 0ull;
  __syncthreads();

  const int wave = tid >> 5, lane = tid & 31;
  const int NT = (VV * 4) / 512;                 // 1000 tiles of 128 floats
  uint32_t lds0 = (uint32_t)(uintptr_t)&stage[0][wave][lane * 4];
  uint32_t lds1 = (uint32_t)(uintptr_t)&stage[1][wave][lane * 4];
  uint64_t gbase = (uint64_t)(uintptr_t)rowp + (uint64_t)lane * 16;

  {
    uint64_t ga = gbase + (uint64_t)wave * 512;
    asm volatile("global_load_async_to_lds_b128 %0, %1, off"
                 :: "v"(lds0), "v"(ga) : "memory");
  }
  for (int t = wave, j = 0; t < NT; t += NWAVES, ++j) {
    int tn = t + NWAVES;
    if (tn < NT) {
      uint32_t ldsNext = ((j + 1) & 1) ? lds1 : lds0;
      uint64_t ga = gbase + (uint64_t)tn * 512;
      asm volatile("global_load_async_to_lds_b128 %0, %1, off"
                   :: "v"(ldsNext), "v"(ga) : "memory");
      asm volatile("s_wait_asynccnt 1" ::: "memory");
    } else {
      asm volatile("s_wait_asynccnt 0" ::: "memory");
    }
    float4 x = *(const float4*)&stage[j & 1][wave][lane * 4];
#pragma unroll
    for (int c = 0; c < 4; ++c) {
      float d = comp4(x, c) * invT - m;
      float e = __expf(d);
      unsigned long long pv = (1ull << CNT_SHIFT) |
                              (unsigned long long)(e * SC1);
      atomicAdd(&hist[bucket1(d)], pv);
    }
  }
  __syncthreads();

  // ---------------- level-1 crossing searches (wave-parallel) -------------
  if (tid < 32) {
    auto wm = [&](int b) { return hist[b] & MASK46; };
    auto wc = [&](int b) { return hist[b] >> CNT_SHIFT; };
    // fixed-point total mass (self-consistent denominator for targets)
    unsigned long long lm = 0ull;
    int base = NB1 - 1 - lane * PL1;
    for (int i = 0; i < PL1; ++i) lm += wm(base - i);
    unsigned long long totalMass = __shfl(waveInclScan(lm), 31, 32);

    unsigned long long tg0 = (unsigned long long)((double)topp2 * (double)totalMass);
    unsigned long long tg1 = (unsigned long long)((double)topp  * (double)totalMass);
    if (tg0 == 0ull) tg0 = 1ull;
    if (tg1 == 0ull) tg1 = 1ull;

    int b; unsigned long long prev;
    bool f0 = waveFindCross(wm, NB1, PL1, tg0, b, prev);
    if (lane == 0) { s_found[0] = f0; if (f0) { s_cb[0] = b; s_cr[0] = tg0 - prev; } }
    bool f1 = waveFindCross(wm, NB1, PL1, tg1, b, prev);
    if (lane == 0) { s_found[1] = f1; if (f1) { s_cb[1] = b; s_cr[1] = tg1 - prev; } }
    bool f2 = waveFindCross(wc, NB1, PL1, (unsigned long long)k, b, prev);
    if (lane == 0) { s_found[2] = f2; if (f2) { s_cb[2] = b; s_ckr = (unsigned long long)k - prev; } }
  }
  __syncthreads();

  // ---------------- P1b: level-2 refinement histograms --------------------
  for (int i = tid; i < NB2; i += NTHREADS) {
    scnt[i] = 0u; subm[0][i] = 0ull; subm[1][i] = 0ull; subm[2][i] = 0ull;
  }
  __syncthreads();
  {
    int b0 = s_found[0] ? s_cb[0] : -1;
    int b1 = s_found[1] ? s_cb[1] : -1;
    int b2 = s_found[2] ? s_cb[2] : -1;
    float lo0 = -32.0f + b0 * W1, lo1 = -32.0f + b1 * W1, lo2 = -32.0f + b2 * W1;
    for (int i = tid; i < VV / 4; i += NTHREADS) {
      float4 x = r4[i];
#pragma unroll
      for (int c = 0; c < 4; ++c) {
        float d = comp4(x, c) * invT - m;
        int b = bucket1(d);
        if (b == b0) atomicAdd(&subm[0][bucket2(d, lo0)],
                               (unsigned long long)(__expf(d) * SC2));
        if (b == b1) atomicAdd(&subm[1][bucket2(d, lo1)],
                               (unsigned long long)(__expf(d) * SC2));
        if (b == b2) atomicAdd(&scnt[bucket2(d, lo2)], 1u);
      }
    }
  }
  __syncthreads();
  if (tid < 32) {
    for (int t = 0; t < 2; ++t) {
      float thr = -3.0e38f;
      if (s_found[t]) {
        float lo = -32.0f + s_cb[t] * W1;
        const unsigned long long* sm = subm[t];
        auto ws = [&](int s) { return sm[s]; };
        int s; unsigned long long prev;
        thr = lo;
        if (waveFindCross(ws, NB2, PL2, s_cr[t] << 6, s, prev)) thr = lo + s * W2;
      }
      if (lane == 0) s_thr[t] = thr;
    }
    {
      float thr = -3.0e38f;
      if (s_found[2]) {
        float lo = -32.0f + s_cb[2] * W1;
        auto ws = [&](int s) { return (unsigned long long)scnt[s]; };
        int s; unsigned long long prev;
        thr = lo;
        if (waveFindCross(ws, NB2, PL2, s_ckr, s, prev)) thr = lo + s * W2;
      }
      if (lane == 0) {
        s_thr[2] = thr;
        float tm = (minp > 0.0f) ? logf(minp) : -3.0e38f;
        c_tp2 = s_thr[0];
        c_ts = fmaxf(fmaxf(s_thr[1], thr), tm);
      }
    }
  }
  __syncthreads();
  const float tp2 = c_tp2, ts = c_ts;

  // ---------------- P2: exact kept-set masses (deterministic) -------------
  float s2p = 0.0f, ssp = 0.0f;
  for (int i = tid; i < VV / 4; i += NTHREADS) {
    float4 x = r4[i];
#pragma unroll
    for (int c = 0; c < 4; ++c) {
      float d = comp4(x, c) * invT - m;
      float e = expf(d);
      if (d >= tp2) s2p += e;
      if (d >= ts) ssp += e;
    }
  }
  const float S2 = blockSum(s2p, red);
  const float Ss = blockSum(ssp, red);

  // ---------------- P3: inverse-CDF crossing ------------------------------
  if (tid < 32) {
    auto wm = [&](int b) { return hist[b] & MASK46; };
    unsigned long long tgt = (unsigned long long)((double)uu * (double)Ss * (double)SC1);
    if (tgt == 0ull) tgt = 1ull;
    int b; unsigned long long prev;
    bool f = waveFindCross(wm, NB1, PL1, tgt, b, prev);
    if (lane == 0) {
      s_found[3] = f;
      if (f) { s_cb[3] = b; s_cr[3] = tgt - prev; }
      else   { s_cb[3] = -1; s_cr[3] = 1ull; }
    }
  }
  __syncthreads();
  if (s_cb[3] < 0) {
    // fp fallback: lowest non-empty bucket (rare; deepest kept tail)
    if (tid < 32) {
      auto wm = [&](int b) { return hist[b] & MASK46; };
      unsigned long long lm = 0ull;
      int base = NB1 - 1 - lane * PL1;
      for (int i = 0; i < PL1; ++i) lm += wm(base - i);
      unsigned long long totalMass = __shfl(waveInclScan(lm), 31, 32);
      int b; unsigned long long prev;
      if (waveFindCross(wm, NB1, PL1, totalMass, b, prev) && lane == 0) {
        s_cb[3] = b; s_cr[3] = totalMass - prev;
      }
    }
    __syncthreads();
  }
  for (int i = tid; i < NB2; i += NTHREADS) subm[0][i] = 0ull;
  __syncthreads();
  const int b3 = (s_cb[3] >= 0) ? s_cb[3] : 0;
  const float lo3 = -32.0f + b3 * W1;
  for (int i = tid; i < VV / 4; i += NTHREADS) {
    float4 x = r4[i];
#pragma unroll
    for (int c = 0; c < 4; ++c) {
      float d = comp4(x, c) * invT - m;
      if (bucket1(d) == b3)
        atomicAdd(&subm[0][bucket2(d, lo3)], (unsigned long long)(__expf(d) * SC2));
    }
  }
  __syncthreads();
  if (tid < 32) {
    auto ws = [&](int s) { return subm[0][s]; };
    int s; unsigned long long prev;
    int sel = 0;
    if (waveFindCross(ws, NB2, PL2, s_cr[3] << 6, s, prev)) sel = s;
    if (lane == 0) { c_s3 = sel; selKey = 0ull; }
  }
  __syncthreads();
  const int s3 = c_s3;

  // ---------------- P4: deterministic element selection -------------------
  for (int i = tid; i < VV / 4; i += NTHREADS) {
    float4 x = r4[i];
#pragma unroll
    for (int c = 0; c < 4; ++c) {
      float d = comp4(x, c) * invT - m;
      if (bucket1(d) == b3 && bucket2(d, lo3) == s3) {
        unsigned ub = __float_as_uint(d);
        unsigned key = (ub & 0x80000000u) ? ~ub : (ub | 0x80000000u);
        unsigned idx = (unsigned)(i * 4 + c);
        unsigned long long kk = ((unsigned long long)key << 32) | (unsigned)(~idx);
        atomicMax(&selKey, kk);
      }
    }
  }
  __syncthreads();

  const float logS2 = logf(S2);
  if (tid == 0) {
    unsigned long long kk = selKey;
    int token = 0; float dtok = 0.0f;
    if (kk != 0ull) {
      token = (int)(~((unsigned)kk));
      unsigned key = (unsigned)(kk >> 32);
      unsigned ub = (key & 0x80000000u) ? (key ^ 0x80000000u) : ~key;
      dtok = __uint_as_float(ub);
    }
    tok_out[row] = token;
    ntlp_out[row] = (dtok >= tp2) ? (dtok - logS2) : FMIN_F32;
  }

  // ---------------- P5: full logprobs output ------------------------------
  float4* o4 = (float4*)(lp_out + (size_t)row * VV);
  for (int i = tid; i < VV / 4; i += NTHREADS) {
    float4 x = r4[i];
    float4 o;
    float d = x.x * invT - m; o.x = (d >= tp2) ? (d - logS2) : FMIN_F32;
    d = x.y * invT - m;       o.y = (d >= tp2) ? (d - logS2) : FMIN_F32;
    d = x.z * invT - m;       o.z = (d >= tp2) ? (d - logS2) : FMIN_F32;
    d = x.w * invT - m;       o.w = (d >= tp2) ? (d - logS2) : FMIN_F32;
    o4[i] = o;
  }
}

extern "C" void kernel_launch(void* const* d_in, const int* in_sizes, int n_in,
                              void* d_out, int out_size, void* d_ws, size_t ws_size,
                              hipStream_t stream) {
  (void)in_sizes; (void)n_in; (void)out_size; (void)d_ws; (void)ws_size;
  const float* logits = (const float*)d_in[0];
  const float* temps  = (const float*)d_in[1];
  const int*   topks  = (const int*)d_in[2];
  const float* topps  = (const float*)d_in[3];
  const float* topps2 = (const float*)d_in[4];
  const float* minps  = (const float*)d_in[5];
  const float* us     = (const float*)d_in[6];
  int*   tok  = (int*)d_out;
  float* lp   = (float*)d_out + BB;
  float* ntlp = (float*)d_out + BB + (size_t)BB * VV;
  hipLaunchKernelGGL(Sampler_77575699300381_kernel, dim3(BB), dim3(NTHREADS), 0, stream,
                     logits, temps, topks, topps, topps2, minps, us, tok, lp, ntlp);
}